// FeatureMapLocation_26749056320399
// MI455X (gfx1250) — compile-verified
//
#include <hip/hip_runtime.h>

// ---------------------------------------------------------------------------
// FeatureMapLocation for MI455X (gfx1250).
// Memory-bound problem (~302 MB traffic @ 23.3 TB/s ~= 13us floor; 50 MFLOP
// of compute is irrelevant). Heat pass uses V_WMMA_F32_16X16X4_F32 so the
// dot products run on the CDNA5 matrix pipe at full fp32 precision; the
// multiply pass is a float4-vectorized stream kernel.
// ---------------------------------------------------------------------------

typedef float v2f __attribute__((ext_vector_type(2)));
typedef float v8f __attribute__((ext_vector_type(8)));

#define NB   64      // batch
#define NC   2048    // channels
#define NHW  192     // 24*8 positions
#define NHW4 48      // NHW / 4
#define NTIL 12      // NHW / 16 M-tiles per sample

// ---------------------------------------------------------------------------
// Kernel 1: heat[b][p] = sum_c W[pids[b]][c] * F[b][c][p], then per-sample
// min/max normalization. One block per sample, 12 waves; wave t owns the
// 16-position tile [16t, 16t+16).
//
// WMMA mapping (V_WMMA_F32_16X16X4_F32, D = A(16x4) * B(4x16) + C):
//   A[m][k] = F[b][k0+k][mbase+m]   (positions on M, channels on K)
//   B[k][n] = w[k0+k]  for all n    (w broadcast across N columns)
// ISA layout (wave32): lane l, half = l>>4, m = l&15:
//   A vgpr0 = A[m][2*half], A vgpr1 = A[m][2*half+1]
//   B vgpr0 = B[2*half][n], B vgpr1 = B[2*half+1][n]   (n = l&15)
//   D vgpr v = D[v + 8*half][n]  -> all N columns identical = heat values.
// ---------------------------------------------------------------------------
__global__ __launch_bounds__(384) void fml_heat_kernel(
    const float* __restrict__ F,        // [NB][NC][NHW]
    const float* __restrict__ Wcls,     // [NUM_IDS][NC]
    const long long* __restrict__ pids, // [NB] (int64 in reference)
    float* __restrict__ heat_out)       // [NB][NHW] normalized
{
    __shared__ float wrow[NC];   // 8 KB
    __shared__ float heat[NHW];  // 768 B

    const int b   = blockIdx.x;
    const int tid = threadIdx.x;

    // Stage this sample's classifier row into LDS.
    const long long pid = pids[b];
    const float* wsrc = Wcls + (size_t)pid * NC;
    for (int i = tid; i < NC; i += 384) wrow[i] = wsrc[i];
    __syncthreads();

    const int wave  = tid >> 5;   // 0..11 -> M tile
    const int lane  = tid & 31;
    const int kh    = lane >> 4;  // which K-pair this half-wave covers
    const int m     = lane & 15;
    const int mbase = wave * 16;

    const float* fbase = F + (size_t)b * NC * NHW + mbase + m;

    v8f d = {0.f, 0.f, 0.f, 0.f, 0.f, 0.f, 0.f, 0.f};

#pragma unroll 4
    for (int k0 = 0; k0 < NC; k0 += 4) {
        const int ka = k0 + 2 * kh;
        v2f a, bm;
        a.x  = fbase[(size_t)ka * NHW];        // F[b][ka  ][mbase+m]
        a.y  = fbase[(size_t)(ka + 1) * NHW];  // F[b][ka+1][mbase+m]
        bm.x = wrow[ka];                       // contiguous pair -> ds b64
        bm.y = wrow[ka + 1];
        // 8 args: (neg_a, A, neg_b, B, c_mod, C, reuse_a, reuse_b)
        d = __builtin_amdgcn_wmma_f32_16x16x4_f32(
                false, a, false, bm, (short)0, d, false, false);
    }

    // All 16 N-columns of D are identical; one lane per half writes its 8 rows.
    if (m == 0) {
#pragma unroll
        for (int v = 0; v < 8; ++v)
            heat[mbase + v + 8 * kh] = d[v];
    }
    __syncthreads();

    // Per-sample min/max, computed redundantly per thread (192 LDS broadcast
    // reads; trivial next to the 100 MB streamed above).
    float mn = heat[0], mx = heat[0];
    for (int i = 1; i < NHW; ++i) {
        const float h = heat[i];
        mn = fminf(mn, h);
        mx = fmaxf(mx, h);
    }

    // Normalize only if max != 0 (reference semantics) and emit to scratch.
    if (tid < NHW) {
        const float h  = heat[tid];
        const float hn = (mx != 0.f) ? (h - mn) / (mx - mn) : h;
        heat_out[b * NHW + tid] = hn;
    }
}

// ---------------------------------------------------------------------------
// Kernel 2: out[b][c][p] = F[b][c][p] * heat[b][p], float4-vectorized.
// This streams the dominant 200 MB; fully coalesced 16B accesses.
// ---------------------------------------------------------------------------
__global__ __launch_bounds__(256) void fml_mul_kernel(
    const float4* __restrict__ F4,     // [NB*NC*NHW4]
    const float*  __restrict__ heat,   // [NB][NHW]
    float4* __restrict__ out4)
{
    const size_t total4 = (size_t)NB * NC * NHW4;
    const size_t i = (size_t)blockIdx.x * blockDim.x + threadIdx.x;
    if (i >= total4) return;

    const size_t per_sample4 = (size_t)NC * NHW4;
    const int b  = (int)(i / per_sample4);
    const int p4 = (int)(i % NHW4);

    const float4 f = F4[i];
    // heat rows are 768 B per sample -> 16B aligned; float4 load is safe.
    const float4 h = ((const float4*)(heat))[b * NHW4 + p4];

    float4 o;
    o.x = f.x * h.x;
    o.y = f.y * h.y;
    o.z = f.z * h.z;
    o.w = f.w * h.w;
    out4[i] = o;
}

extern "C" void kernel_launch(void* const* d_in, const int* in_sizes, int n_in,
                              void* d_out, int out_size, void* d_ws, size_t ws_size,
                              hipStream_t stream) {
    const float*     F    = (const float*)d_in[0];      // [64,2048,24,8] f32
    const float*     Wcls = (const float*)d_in[1];      // [751,2048]     f32
    const long long* pids = (const long long*)d_in[2];  // [64]           i64

    float*  heat = (float*)d_ws;   // 64*192*4 = 48 KB scratch
    float4* out4 = (float4*)d_out;

    fml_heat_kernel<<<NB, 384, 0, stream>>>(F, Wcls, pids, heat);

    const size_t total4 = (size_t)NB * NC * NHW4;       // 6,291,456
    const int    blocks = (int)((total4 + 255) / 256);  // 24,576
    fml_mul_kernel<<<blocks, 256, 0, stream>>>((const float4*)F, heat, out4);
}